// MSDeformableAttention_24524263260372
// MI455X (gfx1250) — compile-verified
//
#include <hip/hip_runtime.h>
#include <hip/hip_bf16.h>
#include <math.h>

typedef __attribute__((ext_vector_type(2))) float v2f;
typedef __attribute__((ext_vector_type(8))) float v8f;

#define EMBED    256
#define HEADS    8
#define LEVELS   4
#define POINTS   4
#define HEAD_DIM 32
#define BS       2
#define LV       13294
#define LQ       13294
#define MROWS    (BS * LQ)                   // 26588
#define TP       (HEADS * LEVELS * POINTS)   // 128
#define NT       4                           // 16x16 n-tiles per wave (16x64)

// ---------------------------------------------------------------------------
// f32 GEMM: C[M,N] = A[M,K] @ W[K,N] + bias[N]; optional per-row zero mask.
// One wave computes a 16x64 tile: the A fragment is loaded once per k-step
// and reused by 4 back-to-back V_WMMA_F32_16X16X4_F32 issues.
// ---------------------------------------------------------------------------
__global__ __launch_bounds__(32) void gemm_f32_wmma(
    const float* __restrict__ A, const float* __restrict__ W,
    const float* __restrict__ bias, const unsigned char* __restrict__ mask,
    float* __restrict__ C, int M, int N, int K)
{
    const int lane = threadIdx.x & 31;
    const int half = lane >> 4;      // 0: K,K+1   1: K+2,K+3
    const int l16  = lane & 15;
    const int mBase = blockIdx.x * 16;
    const int nBase = blockIdx.y * (16 * NT);

    const int mA = min(mBase + l16, M - 1);   // clamp ragged tail for loads
    const float* __restrict__ Arow = A + (size_t)mA * K;

    v8f c[NT] = {};
#pragma unroll 2
    for (int k = 0; k < K; k += 4) {
        const int k0 = k + 2 * half;
        v2f a;
        a.x = Arow[k0 + 0];
        a.y = Arow[k0 + 1];
        const float* __restrict__ Wr0 = W + (size_t)k0 * N + nBase + l16;
        const float* __restrict__ Wr1 = Wr0 + N;
#pragma unroll
        for (int t = 0; t < NT; ++t) {
            v2f bfrag;
            bfrag.x = Wr0[t * 16];
            bfrag.y = Wr1[t * 16];
            c[t] = __builtin_amdgcn_wmma_f32_16x16x4_f32(
                       false, a, false, bfrag, (short)0, c[t], false, false);
        }
    }

#pragma unroll
    for (int t = 0; t < NT; ++t) {
        const int n  = nBase + t * 16 + l16;
        const float bv = bias[n];
#pragma unroll
        for (int j = 0; j < 8; ++j) {
            const int m = mBase + j + 8 * half;   // C/D layout: VGPR j -> M=j / j+8
            if (m < M) {
                float o = c[t][j] + bv;
                if (mask && mask[m]) o = 0.0f;
                C[(size_t)m * N + n] = o;
            }
        }
    }
}

// ---------------------------------------------------------------------------
// Fused softmax + multi-scale deformable bilinear sampling.
// Block = 64 threads = 2 waves per query; 8 lanes per head, each lane owns
// 4 channels -> every corner gather is one b128 load (128B per head-corner).
// ---------------------------------------------------------------------------
__global__ __launch_bounds__(64) void msda_sample(
    const float* __restrict__ vproj,     // [B*LV, 256]
    const float* __restrict__ off_raw,   // [B*LQ, 256]
    const float* __restrict__ attn_raw,  // [B*LQ, 128]
    const float* __restrict__ refpts,    // [B*LQ, LEVELS, 2]
    float* __restrict__ out)             // [B*LQ, 256]
{
    const int row = blockIdx.x;              // b*LQ + q
    const int b   = row / LQ;
    const int h   = threadIdx.x >> 3;        // head 0..7
    const int dg  = threadIdx.x & 7;         // channel group: d = dg*4 .. dg*4+3

    const int Hs[LEVELS]    = {100, 50, 25, 13};
    const int Ws[LEVELS]    = {100, 50, 25, 13};
    const int start[LEVELS] = {0, 10000, 12500, 13125};

    // softmax over the 16 (level,point) logits of this head
    const float* __restrict__ lg = attn_raw + (size_t)row * TP + h * (LEVELS * POINTS);
    float w[LEVELS * POINTS];
    float mx = -1e30f;
#pragma unroll
    for (int i = 0; i < LEVELS * POINTS; ++i) { w[i] = lg[i]; mx = fmaxf(mx, w[i]); }
    float s = 0.0f;
#pragma unroll
    for (int i = 0; i < LEVELS * POINTS; ++i) { w[i] = __expf(w[i] - mx); s += w[i]; }
    const float inv = 1.0f / s;

    const float* __restrict__ offr = off_raw + (size_t)row * (TP * 2);
    const float* __restrict__ refr = refpts + (size_t)row * (LEVELS * 2);
    const float* __restrict__ vb   =
        vproj + ((size_t)b * LV) * EMBED + h * HEAD_DIM + dg * 4;

    float4 acc = make_float4(0.0f, 0.0f, 0.0f, 0.0f);
#pragma unroll
    for (int l = 0; l < LEVELS; ++l) {
        const int   Wl = Ws[l], Hl = Hs[l];
        const float refx = refr[l * 2 + 0];
        const float refy = refr[l * 2 + 1];
#pragma unroll
        for (int p = 0; p < POINTS; ++p) {
            const int   o    = ((h * LEVELS + l) * POINTS + p) * 2;
            const float locx = refx + offr[o + 0] / (float)Wl;   // norm = [W, H]
            const float locy = refy + offr[o + 1] / (float)Hl;
            const float x    = locx * (float)Wl - 0.5f;
            const float y    = locy * (float)Hl - 0.5f;
            const float x0f  = floorf(x), y0f = floorf(y);
            const float lw   = x - x0f,   lh  = y - y0f;
            const int   x0   = (int)x0f,  y0  = (int)y0f;
            const float aw   = w[l * POINTS + p] * inv;

            const float cw[4]  = {(1.0f - lh) * (1.0f - lw), (1.0f - lh) * lw,
                                  lh * (1.0f - lw),           lh * lw};
            const int   dxs[4] = {0, 1, 0, 1};
            const int   dys[4] = {0, 0, 1, 1};
#pragma unroll
            for (int cI = 0; cI < 4; ++cI) {
                const int  xi = x0 + dxs[cI];
                const int  yi = y0 + dys[cI];
                const bool valid = (xi >= 0) & (xi < Wl) & (yi >= 0) & (yi < Hl);
                const int  xc  = min(max(xi, 0), Wl - 1);
                const int  yc  = min(max(yi, 0), Hl - 1);
                const int  idx = start[l] + yc * Wl + xc;
                const float wt = valid ? cw[cI] * aw : 0.0f;
                const float4 g = *(const float4*)(vb + (size_t)idx * EMBED);
                acc.x = fmaf(wt, g.x, acc.x);
                acc.y = fmaf(wt, g.y, acc.y);
                acc.z = fmaf(wt, g.z, acc.z);
                acc.w = fmaf(wt, g.w, acc.w);
            }
        }
    }
    *(float4*)(out + (size_t)row * EMBED + h * HEAD_DIM + dg * 4) = acc;
}

// ---------------------------------------------------------------------------
extern "C" void kernel_launch(void* const* d_in, const int* in_sizes, int n_in,
                              void* d_out, int out_size, void* d_ws, size_t ws_size,
                              hipStream_t stream)
{
    (void)in_sizes; (void)n_in; (void)out_size; (void)ws_size;

    const float*         query  = (const float*)d_in[0];
    const float*         refpts = (const float*)d_in[1];
    const float*         value  = (const float*)d_in[2];
    const unsigned char* vmask  = (const unsigned char*)d_in[3];
    // d_in[4] spatial_shapes: fixed int64 pyramid, hardcoded in kernels
    const float* W_off  = (const float*)d_in[5];
    const float* b_off  = (const float*)d_in[6];
    const float* W_attn = (const float*)d_in[7];
    const float* b_attn = (const float*)d_in[8];
    const float* W_val  = (const float*)d_in[9];
    const float* b_val  = (const float*)d_in[10];
    const float* W_out  = (const float*)d_in[11];
    const float* b_out  = (const float*)d_in[12];
    float*       out    = (float*)d_out;

    float* ws       = (float*)d_ws;
    float* v_proj   = ws;                                   // MROWS*256
    float* off_raw  = v_proj  + (size_t)MROWS * EMBED;      // MROWS*256
    float* attn_raw = off_raw + (size_t)MROWS * EMBED;      // MROWS*128
    float* out_attn = attn_raw + (size_t)MROWS * TP;        // MROWS*256

    const int  Mt = (MROWS + 15) / 16;
    const dim3 blk(32);

    // 1) value projection (+mask)
    gemm_f32_wmma<<<dim3(Mt, EMBED / 64), blk, 0, stream>>>(
        value, W_val, b_val, vmask, v_proj, MROWS, EMBED, EMBED);
    // 2) sampling offsets
    gemm_f32_wmma<<<dim3(Mt, (TP * 2) / 64), blk, 0, stream>>>(
        query, W_off, b_off, nullptr, off_raw, MROWS, TP * 2, EMBED);
    // 3) attention logits
    gemm_f32_wmma<<<dim3(Mt, TP / 64), blk, 0, stream>>>(
        query, W_attn, b_attn, nullptr, attn_raw, MROWS, TP, EMBED);
    // 4) softmax + deformable bilinear sampling
    msda_sample<<<dim3(MROWS), dim3(64), 0, stream>>>(
        v_proj, off_raw, attn_raw, refpts, out_attn);
    // 5) output projection
    gemm_f32_wmma<<<dim3(Mt, EMBED / 64), blk, 0, stream>>>(
        out_attn, W_out, b_out, nullptr, out, MROWS, EMBED, EMBED);
}